// Model_71133248356905
// MI455X (gfx1250) — compile-verified
//
#include <hip/hip_runtime.h>

#define NSTEPS 32
#define ELEMS  4

typedef float v4f __attribute__((ext_vector_type(4)));

// CDNA5 (gfx1250) hardware tanh: single TRANS32 instruction.
__device__ __forceinline__ float htanh(float x) {
#if __has_builtin(__builtin_amdgcn_tanhf)
    return __builtin_amdgcn_tanhf(x);
#else
    float r;
    asm("v_tanh_f32 %0, %1" : "=v"(r) : "v"(x));
    return r;
#endif
}

// One block == one batch row. blockDim.x == tlen/ELEMS (host-enforced),
// so every thread is "full": no modulo, no tail path, no masks.
__global__ __launch_bounds__(256) void node_rk4_rows(
    const float* __restrict__ tbuf,   // (1,T,2) interleaved [t0, t1] pairs
    const float* __restrict__ x,      // (B,T,1)
    const float* __restrict__ W1,     // 2x2 row-major
    const float* __restrict__ b1,     // 2
    const float* __restrict__ W2,     // 1x2
    const float* __restrict__ b2,     // 1
    float* __restrict__ out,          // (B,T,1)
    int tlen)
{
    // Uniform weights -> scalar loads (s_load), live in SGPRs.
    const float w00 = W1[0], w01 = W1[1], w10 = W1[2], w11 = W1[3];
    const float c10 = b1[0], c11 = b1[1];
    const float v0  = W2[0], v1  = W2[1];
    const float c2  = b2[0];

    const long long row = (long long)blockIdx.x * tlen;

    // Grid-stride over columns (exactly one iteration when
    // blockDim.x * ELEMS == tlen, as launched for T=1024).
    #pragma unroll 1
    for (int col0 = threadIdx.x * ELEMS; col0 < tlen; col0 += blockDim.x * ELEMS) {
        const long long base = row + col0;

        // x/out are touched exactly once -> non-temporal; the t vector is
        // re-read by all 16384 blocks at identical addresses -> cached.
        const v4f xv = __builtin_nontemporal_load((const v4f*)(x + base));
        const v4f ta = *(const v4f*)(tbuf + 2 * col0);      // cols col0, col0+1
        const v4f tb = *(const v4f*)(tbuf + 2 * col0 + 4);  // cols col0+2, col0+3

        float y[ELEMS]  = { xv.x, xv.y, xv.z, xv.w };
        float tc[ELEMS] = { ta.x, ta.z, tb.x, tb.z };
        float dt[ELEMS] = { ta.y - ta.x, ta.w - ta.z, tb.y - tb.x, tb.w - tb.z };
        float hd[ELEMS], d6[ELEMS];
        #pragma unroll
        for (int e = 0; e < ELEMS; ++e) {
            dt[e] *= (1.0f / NSTEPS);
            hd[e]  = 0.5f * dt[e];
            d6[e]  = dt[e] * (1.0f / 6.0f);
        }

        // Step loop outer / element loop unrolled inner: 4 independent RK4
        // chains interleave in the schedule, hiding v_tanh_f32 latency.
        #pragma unroll 1
        for (int s = 0; s < NSTEPS; ++s) {
            #pragma unroll
            for (int e = 0; e < ELEMS; ++e) {
                const float t_m = tc[e] + hd[e];
                const float t_e = tc[e] + dt[e];
                // t enters f linearly: fold w*t + b per distinct time point.
                const float a0c = fmaf(w01, tc[e], c10), a1c = fmaf(w11, tc[e], c11);
                const float a0m = fmaf(w01, t_m,  c10), a1m = fmaf(w11, t_m,  c11);
                const float a0e = fmaf(w01, t_e,  c10), a1e = fmaf(w11, t_e,  c11);

                float h0 = htanh(fmaf(w00, y[e], a0c));
                float h1 = htanh(fmaf(w10, y[e], a1c));
                const float k1 = fmaf(v0, h0, fmaf(v1, h1, c2));

                const float y2 = fmaf(hd[e], k1, y[e]);
                h0 = htanh(fmaf(w00, y2, a0m));
                h1 = htanh(fmaf(w10, y2, a1m));
                const float k2 = fmaf(v0, h0, fmaf(v1, h1, c2));

                const float y3 = fmaf(hd[e], k2, y[e]);
                h0 = htanh(fmaf(w00, y3, a0m));
                h1 = htanh(fmaf(w10, y3, a1m));
                const float k3 = fmaf(v0, h0, fmaf(v1, h1, c2));

                const float y4 = fmaf(dt[e], k3, y[e]);
                h0 = htanh(fmaf(w00, y4, a0e));
                h1 = htanh(fmaf(w10, y4, a1e));
                const float k4 = fmaf(v0, h0, fmaf(v1, h1, c2));

                y[e]  = fmaf(d6[e], k1 + 2.0f * (k2 + k3) + k4, y[e]);
                tc[e] = t_e;
            }
        }

        v4f o;
        o.x = y[0]; o.y = y[1]; o.z = y[2]; o.w = y[3];
        __builtin_nontemporal_store(o, (v4f*)(out + base));
    }
}

extern "C" void kernel_launch(void* const* d_in, const int* in_sizes, int n_in,
                              void* d_out, int out_size, void* d_ws, size_t ws_size,
                              hipStream_t stream) {
    (void)n_in; (void)d_ws; (void)ws_size;
    const float* t  = (const float*)d_in[0];  // (1,T,2)
    const float* x  = (const float*)d_in[1];  // (B,T,1)
    const float* W1 = (const float*)d_in[2];
    const float* b1 = (const float*)d_in[3];
    const float* W2 = (const float*)d_in[4];
    const float* b2 = (const float*)d_in[5];
    float* out = (float*)d_out;

    const long long n = (long long)in_sizes[1];   // B*T elements
    const int tlen = in_sizes[0] / 2;             // T (1024)
    const long long rows = n / tlen;              // B (16384)

    // One block per row; each thread handles ELEMS contiguous columns.
    // For T=1024, ELEMS=4 -> block = 256 threads = 8 wave32s; grid = B.
    int block = tlen / ELEMS;                     // requires ELEMS | tlen
    if (block > 1024) block = 1024;               // then grid-stride loop covers rest
    if (block < 32)   block = 32;

    node_rk4_rows<<<dim3((unsigned)rows), dim3(block), 0, stream>>>(
        t, x, W1, b1, W2, b2, out, tlen);
    (void)out_size;
}